// PointCloudVAELoss_52321291600112
// MI455X (gfx1250) — compile-verified
//
#include <hip/hip_runtime.h>
#include <hip/hip_bf16.h>

typedef __attribute__((ext_vector_type(2))) float v2f;
typedef __attribute__((ext_vector_type(8))) float v8f;

#define NP 2048
#define NG 2048
#define BATCH 32
#define LAT 256
#define BIGF 1e18f
#define EPSF 1e-06f
#define FMAXF 0x1.fffffep+127f

// ---------------------------------------------------------------------------
// Init: colmin[b][n] = +inf bits (dist_sq >= 0, so uint ordering == float ordering)
// ---------------------------------------------------------------------------
__global__ __launch_bounds__(256) void init_kernel(unsigned int* __restrict__ colmin, int n) {
    int i = blockIdx.x * 256 + threadIdx.x;
    if (i < n) colmin[i] = 0x7F800000u;
}

// ---------------------------------------------------------------------------
// Main fused distance kernel.
// A' = -2 * pred_xyz (K padded to 4), C preset to pn[m] + tn[n], so one
// v_wmma_f32_16x16x4_f32 yields the finished squared distance tile.
// All column-side data staged in LDS; inner loop = ds_load + WMMA + min/argmin.
// grid = (NP/128, B), block = 256 (8 waves); wave w handles 16 pred rows.
// ---------------------------------------------------------------------------
__global__ __launch_bounds__(256) void dist_kernel(
    const float* __restrict__ preds,    // [B,5,NP]
    const float* __restrict__ target,   // [B,4,NG]
    const float* __restrict__ tmask,    // [B,NG]
    unsigned int* __restrict__ colmin,  // [B,NG] float bits
    float* __restrict__ part_minpred,   // [gridY*gridX]
    float* __restrict__ part_absE)      // [gridY*gridX]
{
    const int b    = blockIdx.y;
    const int wave = threadIdx.x >> 5;
    const int lane = threadIdx.x & 31;
    const int half = lane >> 4;       // 0: lanes 0-15, 1: lanes 16-31
    const int lc   = lane & 15;
    const int m0   = blockIdx.x * 128 + wave * 16;

    const float* px = preds + ((size_t)b * 5 + 0) * NP;
    const float* py = preds + ((size_t)b * 5 + 1) * NP;
    const float* pz = preds + ((size_t)b * 5 + 2) * NP;
    const float* pE = preds + ((size_t)b * 5 + 3) * NP;
    const float* tx = target + ((size_t)b * 4 + 0) * NG;
    const float* ty = target + ((size_t)b * 4 + 1) * NG;
    const float* tz = target + ((size_t)b * 4 + 2) * NG;
    const float* tE = target + ((size_t)b * 4 + 3) * NG;
    const float* mk = tmask + (size_t)b * NG;

    __shared__ v2f          s_bops[2][NG];   // [0]=(tx,ty)  [1]=(tz,0)   32 KB
    __shared__ float        s_tn[NG];        // ||t||^2                    8 KB
    __shared__ float        s_big[NG];       // mask ? 0 : BIG             8 KB
    __shared__ unsigned int s_colmin[NG];    // block-local column mins    8 KB
    __shared__ float        s_wsum[8];
    __shared__ float        s_wabs[8];

    // Stage all target-column data into LDS once per block.
    for (int i = threadIdx.x; i < NG; i += 256) {
        float TX = tx[i], TY = ty[i], TZ = tz[i];
        v2f b0; b0.x = TX; b0.y = TY;
        v2f b1; b1.x = TZ; b1.y = 0.0f;
        s_bops[0][i] = b0;
        s_bops[1][i] = b1;
        s_tn[i]  = TX * TX + TY * TY + TZ * TZ;
        s_big[i] = (mk[i] != 0.0f) ? 0.0f : BIGF;
        s_colmin[i] = 0x7F800000u;
        __builtin_prefetch(&tE[i], 0, 3);   // gfx1250 global_prefetch for the E gather
    }
    __syncthreads();

    // A' operand: 16x4 f32 = -2*pred; lanes 0-15 hold (K0,K1), lanes 16-31 (K2,K3)
    v2f a;
    {
        int m = m0 + lc;
        if (half == 0) { a.x = -2.0f * px[m]; a.y = -2.0f * py[m]; }
        else           { a.x = -2.0f * pz[m]; a.y = 0.0f; }
    }

    // pn for the 8 output rows this lane sees: M = r + 8*half (C/D layout)
    float pnv[8];
#pragma unroll
    for (int r = 0; r < 8; ++r) {
        int m = m0 + r + 8 * half;
        float X = px[m], Y = py[m], Z = pz[m];
        pnv[r] = X * X + Y * Y + Z * Z;
    }

    float runmin[8];
    int   runarg[8];
#pragma unroll
    for (int r = 0; r < 8; ++r) { runmin[r] = BIGF; runarg[r] = 0; }

#pragma unroll 2
    for (int j = 0; j < NG / 16; ++j) {
        const int ncol = j * 16 + lc;

        v2f   bm   = s_bops[half][ncol];
        float tnv  = s_tn[ncol];
        float bigv = s_big[ncol];

        // C[m][n] = pn[m] + tn[n]; WMMA adds -2*dot -> finished distance.
        v8f c;
#pragma unroll
        for (int r = 0; r < 8; ++r) c[r] = pnv[r] + tnv;

        c = __builtin_amdgcn_wmma_f32_16x16x4_f32(
                /*neg_a=*/false, a, /*neg_b=*/false, bm,
                /*c_mod=*/(short)0, c, /*reuse_a=*/false, /*reuse_b=*/false);

        // med3(x, 0, +FLT_MAX) == max(x, 0) for finite x: single v_med3_num_f32,
        // avoids the NaN-canonicalize double-max that fmaxf generates.
        float d0 = __builtin_amdgcn_fmed3f(c[0], 0.0f, FMAXF);
        float d1 = __builtin_amdgcn_fmed3f(c[1], 0.0f, FMAXF);
        float d2 = __builtin_amdgcn_fmed3f(c[2], 0.0f, FMAXF);
        float d3 = __builtin_amdgcn_fmed3f(c[3], 0.0f, FMAXF);
        float d4 = __builtin_amdgcn_fmed3f(c[4], 0.0f, FMAXF);
        float d5 = __builtin_amdgcn_fmed3f(c[5], 0.0f, FMAXF);
        float d6 = __builtin_amdgcn_fmed3f(c[6], 0.0f, FMAXF);
        float d7 = __builtin_amdgcn_fmed3f(c[7], 0.0f, FMAXF);
        float dv[8] = { d0, d1, d2, d3, d4, d5, d6, d7 };

        // column min across the 8 rows this lane holds (min3 tree)
        float cmin = __builtin_amdgcn_fmed3f(
            __builtin_amdgcn_fmed3f(d0 < d1 ? d0 : d1, d2 < d3 ? d2 : d3, FMAXF),
            __builtin_amdgcn_fmed3f(d4 < d5 ? d4 : d5, d6 < d7 ? d6 : d7, FMAXF),
            FMAXF);

#pragma unroll
        for (int r = 0; r < 8; ++r) {
            float dm = dv[r] + bigv;    // exact: d + 1e18 rounds to 1e18
            if (dm < runmin[r]) { runmin[r] = dm; runarg[r] = ncol; }
        }
        atomicMin(&s_colmin[ncol], __float_as_uint(cmin));
    }

    // Cross-lane row-min reduce within each 16-lane half (xor of bits 0..3 stays in half)
    float rsum = 0.0f, esum = 0.0f;
#pragma unroll
    for (int r = 0; r < 8; ++r) {
        float mn = runmin[r];
        int   ar = runarg[r];
#pragma unroll
        for (int s = 1; s < 16; s <<= 1) {
            float om = __shfl_xor(mn, s, 32);
            int   oa = __shfl_xor(ar, s, 32);
            if (om < mn || (om == mn && oa < ar)) { mn = om; ar = oa; }
        }
        if (lc == 0) {
            int m = m0 + r + 8 * half;
            rsum += mn;
            esum += fabsf(pE[m] - tE[ar]);
        }
    }
    // combine the two halves (only lanes 0 and 16 are nonzero)
    rsum += __shfl_xor(rsum, 16, 32);
    esum += __shfl_xor(esum, 16, 32);
    if (lane == 0) { s_wsum[wave] = rsum; s_wabs[wave] = esum; }
    __syncthreads();

    if (threadIdx.x == 0) {
        float a0 = 0.0f, a1 = 0.0f;
        for (int w = 0; w < 8; ++w) { a0 += s_wsum[w]; a1 += s_wabs[w]; }
        int bid = blockIdx.y * gridDim.x + blockIdx.x;
        part_minpred[bid] = a0;
        part_absE[bid]    = a1;
    }

    // Flush block's column mins (atomic min on uint bits: deterministic)
    for (int i = threadIdx.x; i < NG; i += 256)
        atomicMin(&colmin[(size_t)b * NG + i], s_colmin[i]);
}

// ---------------------------------------------------------------------------
// Per-batch streaming reductions (deterministic LDS tree)
// out_b[b*6 + {0:ct,1:masksum,2:nhits,3:etotal,4:entropy,5:kld}]
// ---------------------------------------------------------------------------
__global__ __launch_bounds__(256) void batch_kernel(
    const float* __restrict__ preds,
    const float* __restrict__ tmask,
    const unsigned int* __restrict__ colmin,
    const float* __restrict__ mu,
    const float* __restrict__ logvar,
    float* __restrict__ out_b)
{
    const int b   = blockIdx.x;
    const int tid = threadIdx.x;

    const float* pE = preds + ((size_t)b * 5 + 3) * NP;
    const float* ph = preds + ((size_t)b * 5 + 4) * NP;
    const float* mk = tmask + (size_t)b * NG;

    float ct = 0.0f, ms = 0.0f, nh = 0.0f, et = 0.0f, en = 0.0f, kl = 0.0f;
    for (int i = tid; i < NG; i += 256) {
        float mv = mk[i];
        ct += __uint_as_float(colmin[(size_t)b * NG + i]) * mv;
        ms += mv;
        float h = ph[i];
        nh += h;
        et += pE[i] * h;
        en += h * logf(h + EPSF) + (1.0f - h) * logf(1.0f - h + EPSF);
    }
    if (tid < LAT) {
        float m  = mu[(size_t)b * LAT + tid];
        float lv = logvar[(size_t)b * LAT + tid];
        kl = 1.0f + lv - m * m - expf(lv);
    }

    __shared__ float sred[6][256];
    sred[0][tid] = ct; sred[1][tid] = ms; sred[2][tid] = nh;
    sred[3][tid] = et; sred[4][tid] = en; sred[5][tid] = kl;
    __syncthreads();
    for (int off = 128; off > 0; off >>= 1) {
        if (tid < off)
            for (int k = 0; k < 6; ++k) sred[k][tid] += sred[k][tid + off];
        __syncthreads();
    }
    if (tid == 0)
        for (int k = 0; k < 6; ++k) out_b[b * 6 + k] = sred[k][0];
}

// ---------------------------------------------------------------------------
// Final combine (single thread, fixed order => deterministic)
// ---------------------------------------------------------------------------
__global__ void final_kernel(
    const float* __restrict__ part_minpred,
    const float* __restrict__ part_absE,
    int nparts,
    const float* __restrict__ out_b,
    const float* __restrict__ e_init,
    const float* __restrict__ kl_weight,
    float* __restrict__ out)
{
    if (threadIdx.x != 0 || blockIdx.x != 0) return;

    float smp = 0.0f, sae = 0.0f;
    for (int i = 0; i < nparts; ++i) { smp += part_minpred[i]; sae += part_absE[i]; }

    float ct = 0.0f, mstot = 0.0f, hit = 0.0f, ent = 0.0f, ge = 0.0f, kl = 0.0f;
    for (int b = 0; b < BATCH; ++b) {
        ct += out_b[b * 6 + 0];
        float msb = out_b[b * 6 + 1];
        mstot += msb;
        float d = out_b[b * 6 + 2] - msb;          // n_hits_pred - n_hits_target
        hit += d * d;
        float de = out_b[b * 6 + 3] - e_init[b];   // total_pred_E - e_init
        ge += de * de;
        ent += out_b[b * 6 + 4];
        kl  += out_b[b * 6 + 5];
    }

    const float Bf = (float)BATCH, Nf = (float)NP;
    float loss_chamfer     = ct / mstot + smp / (Bf * Nf);
    float loss_localE      = sae / (Bf * Nf);
    float loss_hit_count   = hit / Bf;
    float loss_hit_entropy = -ent / (Bf * Nf);
    float loss_globalE     = ge / Bf;
    float loss_kld         = -0.5f * kl / Bf;

    float c_chamf = loss_chamfer * 0.001f;      // LAMBDA_CHAMFER
    float c_loc   = loss_localE;
    float c_glob  = 10.0f * loss_globalE;       // LAMBDA_E_SUM
    float c_hit   = 20.0f * loss_hit_count;     // LAMBDA_HIT
    float c_kld   = kl_weight[0] * loss_kld;
    float c_entr  = 0.1f * loss_hit_entropy;    // LAMBDA_HIT_ENTROPY

    out[0] = c_chamf + c_loc + c_kld + c_glob + c_hit + c_entr;
    out[1] = c_chamf;
    out[2] = c_loc;
    out[3] = c_glob;
    out[4] = c_hit;
    out[5] = c_kld;
}

// ---------------------------------------------------------------------------
extern "C" void kernel_launch(void* const* d_in, const int* in_sizes, int n_in,
                              void* d_out, int out_size, void* d_ws, size_t ws_size,
                              hipStream_t stream) {
    const float* preds     = (const float*)d_in[0];  // [32,5,2048]
    const float* target    = (const float*)d_in[1];  // [32,4,2048]
    const float* tmask     = (const float*)d_in[2];  // [32,2048]
    const float* mu        = (const float*)d_in[3];  // [32,256]
    const float* logvar    = (const float*)d_in[4];  // [32,256]
    const float* e_init    = (const float*)d_in[5];  // [32]
    const float* kl_weight = (const float*)d_in[6];  // [1]
    float* out = (float*)d_out;                      // 6 floats

    // workspace layout
    char* base = (char*)d_ws;
    unsigned int* colmin = (unsigned int*)base;                       // 32*2048 u32
    float* part_minpred  = (float*)(base + (size_t)BATCH * NG * 4);   // 512 f32
    float* part_absE     = part_minpred + 512;                        // 512 f32
    float* out_b         = part_absE + 512;                           // 32*6 f32

    const int ncm = BATCH * NG;
    init_kernel<<<(ncm + 255) / 256, 256, 0, stream>>>(colmin, ncm);

    dim3 grid(NP / 128, BATCH);
    dist_kernel<<<grid, 256, 0, stream>>>(preds, target, tmask, colmin,
                                          part_minpred, part_absE);

    batch_kernel<<<BATCH, 256, 0, stream>>>(preds, tmask, colmin, mu, logvar, out_b);

    final_kernel<<<1, 64, 0, stream>>>(part_minpred, part_absE, (NP / 128) * BATCH,
                                       out_b, e_init, kl_weight, out);
}